// HOG_42236708389556
// MI455X (gfx1250) — compile-verified
//
#include <hip/hip_runtime.h>
#include <stdint.h>

// HOG on MI455X (gfx1250): TDM tensor_load_to_lds for data movement,
// LDS ds_add_f32 atomics for the data-dependent bin scatter,
// V_WMMA_F32_16X16X4_F32 (exact f32) for the 8-row cell pooling.

#define HH    224
#define WW    224
#define CCH   3
#define CELLW 8
#define BINS  9
#define CX_N  28          // WW/CELLW
#define CY_N  28          // HH/CELLW
#define NPAD  256         // padded histogram width: 28*9 = 252 -> 256
#define SROWS 10          // 8 output rows + 2 halo rows
#define SCOLS (WW + 2)    // zero-padded columns

typedef float        v2f __attribute__((ext_vector_type(2)));
typedef float        v8f __attribute__((ext_vector_type(8)));
typedef unsigned int v4u __attribute__((ext_vector_type(4)));
typedef int          v8i __attribute__((ext_vector_type(8)));
typedef int          v4i __attribute__((ext_vector_type(4)));

__global__ __launch_bounds__(256)
void hog_tdm_wmma_kernel(const float* __restrict__ x, float* __restrict__ out) {
  __shared__ float raw[CCH * SROWS * WW];   // TDM destination: [c][row][x] packed
  __shared__ float sbuf[SROWS * SCOLS];     // channel-summed, zero-padded stencil tile
  __shared__ float hrow[CELLW * NPAD];      // per-pixel-row (cell,bin) histograms

  const int tid = threadIdx.x;
  const int cy  = blockIdx.x;   // cell row 0..27
  const int img = blockIdx.y;   // image 0..N-1

  const int y0         = cy * CELLW - 1;           // first (halo) image row
  const int yStart     = (y0 < 0) ? 0 : y0;        // clamp top
  const int rStart     = yStart - y0;              // 1 iff cy==0
  const int tileRows   = SROWS - rStart;           // rows actually DMA'd
  const int chanStride = tileRows * WW;            // LDS elements per channel

  // ---------------- Phase 0: TDM async tensor load (Global -> LDS) ----------
#if __has_builtin(__builtin_amdgcn_tensor_load_to_lds)
  if (tid < 32) {  // one wave issues the DMA; TENSORcnt is per-wave
    const uint64_t gaddr = (uint64_t)(uintptr_t)
        (x + (size_t)img * CCH * HH * WW + (size_t)yStart * WW);
    const uint32_t ldsa  = (uint32_t)(uintptr_t)(&raw[0]); // low 32 bits = LDS offset
    const uint32_t tdim0 = WW;                      // tensor length dim0 (cols)
    const uint32_t tdim1 = (uint32_t)(HH - yStart); // rows left: OOB rows zero-fill
    const uint64_t d0s   = WW;                      // row stride (elements)
    const uint64_t d1s   = (uint64_t)HH * WW;       // channel stride (elements)

    v4u g0;
    g0.x = 1u;                                            // count=1, user mode
    g0.y = ldsa;                                          // lds_addr (bytes)
    g0.z = (uint32_t)(gaddr & 0xFFFFFFFFull);             // global_addr[31:0]
    g0.w = (uint32_t)((gaddr >> 32) & 0x1FFFFFFull)       // global_addr[56:32]
         | (2u << 30);                                    // type = 2 ("image")

    v8i g1;
    g1[0] = (int)(2u << 16);                              // data_size = 2 (4 bytes)
    g1[1] = (int)((tdim0 & 0xFFFFu) << 16);               // tensor_dim0[15:0]
    g1[2] = (int)((tdim0 >> 16) | ((tdim1 & 0xFFFFu) << 16)); // dim0 hi | dim1 lo
    g1[3] = (int)((tdim1 >> 16) | ((uint32_t)WW << 16));  // dim1 hi | tile_dim0=224
    g1[4] = (int)(((uint32_t)tileRows) | ((uint32_t)CCH << 16)); // tile_dim1|tile_dim2
    g1[5] = (int)(uint32_t)(d0s & 0xFFFFFFFFull);         // dim0_stride[31:0]
    g1[6] = (int)((uint32_t)((d0s >> 32) & 0xFFFFull)     // dim0_stride[47:32]
         | ((uint32_t)(d1s & 0xFFFFull) << 16));          // dim1_stride[15:0]
    g1[7] = (int)(uint32_t)((d1s >> 16) & 0xFFFFFFFFull); // dim1_stride[47:16]

    v4i g2;
    g2[0] = CCH;  // tensor_dim2 = 3 channels
    g2[1] = 0;    // tensor_dim3 (unused), iterate disabled
    g2[2] = 0;    // tensor_dim2_stride[31:0] (unused, 3D tile only)
    g2[3] = 0;    // dim2_stride hi | tile_dim3 = 0
    v4i g3 = {0, 0, 0, 0};

#if defined(__clang_major__) && (__clang_major__ >= 23)
    v8i g4 = {0, 0, 0, 0, 0, 0, 0, 0};
    __builtin_amdgcn_tensor_load_to_lds(g0, g1, g2, g3, g4, 0);
#else
    __builtin_amdgcn_tensor_load_to_lds(g0, g1, g2, g3, 0);
#endif
    __builtin_amdgcn_s_wait_tensorcnt(0);
  }
#else
  // Fallback: cooperative Global -> LDS copy matching the TDM layout.
  for (int i = tid; i < CCH * chanStride; i += 256) {
    const int c   = i / chanStride;
    const int rem = i - c * chanStride;
    const int rr  = rem / WW;
    const int xw  = rem - rr * WW;
    const int y   = yStart + rr;
    raw[i] = (y < HH)
        ? x[((size_t)img * CCH + c) * (size_t)(HH * WW) + (size_t)y * WW + xw]
        : 0.0f;
  }
#endif

  // Zero the histogram region (independent of the DMA target region).
  for (int i = tid; i < CELLW * NPAD; i += 256) hrow[i] = 0.0f;
  __syncthreads();  // raw[] ready (TDM done in wave0 before its barrier arrive)

  // ---------------- Phase 1: channel-sum into zero-padded stencil tile ------
  for (int i = tid; i < SROWS * SCOLS; i += 256) {
    const int r  = i / SCOLS;
    const int xx = i - r * SCOLS;
    const int xw = xx - 1;
    float v = 0.0f;
    if (xw >= 0 && xw < WW && r >= rStart) {  // r<rStart: missing top halo row -> 0
      const int rr = r - rStart;
      v = raw[0 * chanStride + rr * WW + xw]
        + raw[1 * chanStride + rr * WW + xw]
        + raw[2 * chanStride + rr * WW + xw];
    }
    sbuf[i] = v;
  }
  __syncthreads();

  // ---------------- Phase 2: Sobel + binning + LDS-atomic row histograms ----
  if (tid < WW) {
    const int xw  = tid;
    const int xx  = xw + 1;
    const int cxb = (xw >> 3) * BINS;  // cell-column base in histogram
    for (int rr = 0; rr < CELLW; ++rr) {
      const int r = rr + 1;
      const float* s0 = &sbuf[(r - 1) * SCOLS];
      const float* s1 = &sbuf[r * SCOLS];
      const float* s2 = &sbuf[(r + 1) * SCOLS];
      const float gx = (s0[xx + 1] - s0[xx - 1])
                     + 2.0f * (s1[xx + 1] - s1[xx - 1])
                     + (s2[xx + 1] - s2[xx - 1]);
      const float gy = (s2[xx - 1] + 2.0f * s2[xx] + s2[xx + 1])
                     - (s0[xx - 1] + 2.0f * s0[xx] + s0[xx + 1]);
      const float mag = sqrtf(gx * gx + gy * gy);
      const float ang = fabsf(atan2f(gx, gy));
      // Match reference rounding: floor((ang / float(pi)) * 8)
      int bin = (int)floorf((ang / 3.14159274101257324f) * 8.0f);
      bin = bin < 0 ? 0 : (bin > (BINS - 1) ? (BINS - 1) : bin);
      atomicAdd(&hrow[rr * NPAD + cxb + bin], mag);  // ds_add_f32
    }
  }
  __syncthreads();

  // ---------------- Phase 3: 8-row pooling via f32 WMMA ----------------------
  // hog[n] = sum_{k=0..7} hrow[k][n].  A = all-ones 16x4 f32, so
  // D[m,n] = sum_k B[k,n] regardless of the hardware's slot->k permutation;
  // only the N = lane%16 mapping (fixed by the C/D layout) is relied upon.
#if __has_builtin(__builtin_amdgcn_wmma_f32_16x16x4_f32)
  {
    const int wave = tid >> 5;
    const int lane = tid & 31;
    const int nlo  = lane & 15;
    const int kb   = (lane < 16) ? 0 : 2;  // two K slots per lane half
    for (int t = wave * 2; t < wave * 2 + 2; ++t) {  // 16 N-tiles over 8 waves
      const int n = t * 16 + nlo;
      v2f a; a.x = 1.0f; a.y = 1.0f;
      v8f c = {};
      v2f b0; b0.x = hrow[(kb + 0) * NPAD + n];  // rows {0,1}/{2,3}
              b0.y = hrow[(kb + 1) * NPAD + n];
      c = __builtin_amdgcn_wmma_f32_16x16x4_f32(false, a, false, b0,
                                                (short)0, c, false, false);
      v2f b1; b1.x = hrow[(kb + 4) * NPAD + n];  // rows {4,5}/{6,7}
              b1.y = hrow[(kb + 5) * NPAD + n];
      c = __builtin_amdgcn_wmma_f32_16x16x4_f32(false, a, false, b1,
                                                (short)0, c, false, false);
      if (lane < 16 && n < CX_N * BINS) {
        const int cx  = n / BINS;
        const int bin = n - cx * BINS;
        out[(((size_t)img * BINS + bin) * CY_N + cy) * CX_N + cx] = c[0];
      }
    }
  }
#else
  for (int n = tid; n < CX_N * BINS; n += 256) {
    float acc = 0.0f;
    for (int rr = 0; rr < CELLW; ++rr) acc += hrow[rr * NPAD + n];
    const int cx  = n / BINS;
    const int bin = n - cx * BINS;
    out[(((size_t)img * BINS + bin) * CY_N + cy) * CX_N + cx] = acc;
  }
#endif
}

extern "C" void kernel_launch(void* const* d_in, const int* in_sizes, int n_in,
                              void* d_out, int out_size, void* d_ws, size_t ws_size,
                              hipStream_t stream) {
  (void)n_in; (void)out_size; (void)d_ws; (void)ws_size;
  const float* x = (const float*)d_in[0];
  float* out = (float*)d_out;
  const int N = in_sizes[0] / (CCH * HH * WW);  // 64
  dim3 grid(CY_N, N);   // blockIdx.x = cell row, blockIdx.y = image
  dim3 block(256);      // 8 wave32s
  hog_tdm_wmma_kernel<<<grid, block, 0, stream>>>(x, out);
}